// FullAttention_21698174779448
// MI455X (gfx1250) — compile-verified
//
#include <hip/hip_runtime.h>

// Problem dims (fixed by the reference)
#define B_    16
#define L1_   1024
#define L2_   1024
#define FULL_ 1536
#define H_    768
#define NL_   4
#define D_    192   // H / NL

typedef float v2f __attribute__((ext_vector_type(2)));
typedef float v8f __attribute__((ext_vector_type(8)));

// Load a 16x4 fp32 WMMA A/B fragment from an LDS tile stored row-major.
// ISA layout (16x4 A, and 4x16 B by symmetry): lane L holds row (L&15),
// k-pair base 2*(L>>4), two consecutive k values in 2 VGPRs -> float2.
__device__ __forceinline__ v2f lds_frag(const float* base, int rowBase, int col, int stride) {
  const int lane = threadIdx.x & 31;
  const float* p = base + (rowBase + (lane & 15)) * stride + col + ((lane >> 4) << 1);
  return *(const v2f*)p;
}

__device__ __forceinline__ v8f wmma_f32(v2f a, v2f b, v8f c) {
  // (neg_a, A, neg_b, B, c_mod, C, reuse_a, reuse_b)
  return __builtin_amdgcn_wmma_f32_16x16x4_f32(false, a, false, b, (short)0, c, false, false);
}

// ---------------------------------------------------------------------------
// Kernel 1: Y[r,h] = (scale?) * relu( sum_f X[r,f] * W[h,f] )
// 128 threads / 4 waves per block; block tile 128(r) x 64(h); each wave owns
// 2 M-tiles x 4 N-tiles (8 v8f accumulators) so each k-step is 6 LDS fragment
// loads feeding 8 WMMAs. K staged 32 wide in LDS (stride 36: conflict-free
// fragment reads, float2-aligned).
// ---------------------------------------------------------------------------
__global__ __launch_bounds__(128) void key_gemm_relu(
    const float* __restrict__ X, const float* __restrict__ W,
    const float* __restrict__ scale, int useScale, float* __restrict__ Y) {
  __shared__ float Xs[128 * 36];
  __shared__ float Ws[64 * 36];

  const int hb = blockIdx.x % (H_ / 64);
  const int rb = blockIdx.x / (H_ / 64);
  const int r0 = rb * 128, h0 = hb * 64;
  const int tid  = threadIdx.x;
  const int lane = tid & 31;
  const int wm   = (tid >> 5) * 32;  // wave's 32-row strip (2 M-tiles)

  v8f acc[2][4] = {};

  // Coalesced staging maps: X 128x32 (4 passes of 32 rows), W 64x32.
  const int xrow = tid >> 2;        // 0..31
  const int xcol = (tid & 3) * 8;   // 0,8,16,24
  const int wrow = tid >> 1;        // 0..63
  const int wcol = (tid & 1) * 16;  // 0,16

  for (int f0 = 0; f0 < FULL_; f0 += 32) {
    __syncthreads();
#pragma unroll
    for (int p = 0; p < 4; ++p) {
      const int row = p * 32 + xrow;
      const float* src = X + (size_t)(r0 + row) * FULL_ + f0 + xcol;
      float4 v0 = *(const float4*)(src);
      float4 v1 = *(const float4*)(src + 4);
      float* dst = &Xs[row * 36 + xcol];
      *(v2f*)(dst + 0) = (v2f){v0.x, v0.y};
      *(v2f*)(dst + 2) = (v2f){v0.z, v0.w};
      *(v2f*)(dst + 4) = (v2f){v1.x, v1.y};
      *(v2f*)(dst + 6) = (v2f){v1.z, v1.w};
    }
#pragma unroll
    for (int j = 0; j < 4; ++j) {
      float4 wv = *(const float4*)(W + (size_t)(h0 + wrow) * FULL_ + f0 + wcol + 4 * j);
      float* dst = &Ws[wrow * 36 + wcol + 4 * j];
      *(v2f*)(dst + 0) = (v2f){wv.x, wv.y};
      *(v2f*)(dst + 2) = (v2f){wv.z, wv.w};
    }
    // Hint the next K-panel toward the caches while we compute this one.
    if (f0 + 32 < FULL_) {
      __builtin_prefetch(X + (size_t)(r0 + xrow) * FULL_ + f0 + 32 + xcol, 0, 0);
      __builtin_prefetch(W + (size_t)(h0 + wrow) * FULL_ + f0 + 32 + wcol, 0, 0);
    }
    __syncthreads();
#pragma unroll
    for (int kk = 0; kk < 8; ++kk) {
      v2f a0 = lds_frag(Xs, wm,      kk * 4, 36);
      v2f a1 = lds_frag(Xs, wm + 16, kk * 4, 36);
#pragma unroll
      for (int nt = 0; nt < 4; ++nt) {
        v2f b = lds_frag(Ws, nt * 16, kk * 4, 36);
        acc[0][nt] = wmma_f32(a0, b, acc[0][nt]);
        acc[1][nt] = wmma_f32(a1, b, acc[1][nt]);
      }
    }
  }

  // Epilogue: D layout -> lane holds col (lane&15), rows v + 8*(lane>>4)
  const int half = lane >> 4, col = lane & 15;
#pragma unroll
  for (int mt = 0; mt < 2; ++mt) {
#pragma unroll
    for (int nt = 0; nt < 4; ++nt) {
      const int h = h0 + nt * 16 + col;
      const float sc = useScale ? scale[h] : 1.0f;
#pragma unroll
      for (int v = 0; v < 8; ++v) {
        const float val = fmaxf(acc[mt][nt][v], 0.0f) * sc;
        const int r = r0 + wm + mt * 16 + v + 8 * half;
        Y[(size_t)r * H_ + h] = val;
      }
    }
  }
}

// ---------------------------------------------------------------------------
// Kernel 2: per (b, head, 16-q-row tile) flash-style attention, 1 wave/block.
// Pass 1: streaming row max / sum(exp). Pass 2: recompute S, P via LDS
// layout conversion (D-layout -> A-layout), alpha @ V with 12 n-tiles (D=192).
// ---------------------------------------------------------------------------
__global__ __launch_bounds__(32) void attn_kernel(
    const float* __restrict__ Qk, const float* __restrict__ Kk,
    const float* __restrict__ Vx, const unsigned char* __restrict__ mask,
    float* __restrict__ out) {
  __shared__ float Qs[16 * 196];
  __shared__ float Ks[16 * 196];
  __shared__ float Vs[16 * 196];
  __shared__ float Ps[16 * 20];

  const int idx = blockIdx.x;
  const int b  = idx >> 8;          // / (NL * L1/16)
  const int n  = (idx >> 6) & 3;    // head
  const int q0 = (idx & 63) << 4;   // q-tile origin
  const int lane = threadIdx.x;
  const int half = lane >> 4, coln = lane & 15;

  // Stage Q tile [16 x 192] once.
  {
    const float* src = Qk + ((size_t)b * L1_ + q0) * H_ + n * D_;
    for (int i = lane; i < 768; i += 32) {   // 768 float4s
      const int row = i / 48;
      const int col = (i % 48) * 4;
      *(float4*)&Qs[row * 196 + col] = *(const float4*)(src + (size_t)row * H_ + col);
    }
  }

  float m[8], l[8];
#pragma unroll
  for (int v = 0; v < 8; ++v) { m[v] = -3.0e38f; l[v] = 0.0f; }

  const float* kbase = Kk + (size_t)b * L2_ * H_ + n * D_;
  const float* vbase = Vx + (size_t)b * L2_ * H_ + n * D_;
  const unsigned char* mb = mask + (size_t)b * L2_;

  // ---------------- Pass 1: row max + row sum(exp) ----------------
  for (int kt = 0; kt < L2_ / 16; ++kt) {
    __syncthreads();
    const float* src = kbase + (size_t)kt * 16 * H_;
    for (int i = lane; i < 768; i += 32) {
      const int row = i / 48, col = (i % 48) * 4;
      *(float4*)&Ks[row * 196 + col] = *(const float4*)(src + (size_t)row * H_ + col);
    }
    // L2-resident K: hint the next tile.
    if (kt + 1 < L2_ / 16)
      __builtin_prefetch(src + (size_t)16 * H_ + (lane >> 1) * H_ + (lane & 1) * 96, 0, 0);
    __syncthreads();

    v8f s = {};
#pragma unroll 4
    for (int dk = 0; dk < D_ / 4; ++dk) {
      v2f a  = lds_frag(Qs, 0, dk * 4, 196);
      v2f bf = lds_frag(Ks, 0, dk * 4, 196);
      s = wmma_f32(a, bf, s);
    }
    const bool msk = mb[kt * 16 + coln] != 0;
#pragma unroll
    for (int v = 0; v < 8; ++v) {
      float sv = msk ? -__builtin_inff() : s[v];
      float rm = sv;                                   // reduce over 16-lane half
      rm = fmaxf(rm, __shfl_xor(rm, 1));
      rm = fmaxf(rm, __shfl_xor(rm, 2));
      rm = fmaxf(rm, __shfl_xor(rm, 4));
      rm = fmaxf(rm, __shfl_xor(rm, 8));
      const float mn = fmaxf(m[v], rm);
      float e = __expf(sv - mn);                       // -inf -> 0
      e += __shfl_xor(e, 1); e += __shfl_xor(e, 2);
      e += __shfl_xor(e, 4); e += __shfl_xor(e, 8);
      l[v] = l[v] * __expf(m[v] - mn) + e;
      m[v] = mn;
    }
  }

  // ---------------- Pass 2: recompute S, accumulate alpha @ V ----------------
  v8f acc[12] = {};
  for (int kt = 0; kt < L2_ / 16; ++kt) {
    __syncthreads();
    const float* srcK = kbase + (size_t)kt * 16 * H_;
    const float* srcV = vbase + (size_t)kt * 16 * H_;
    for (int i = lane; i < 768; i += 32) {
      const int row = i / 48, col = (i % 48) * 4;
      *(float4*)&Ks[row * 196 + col] = *(const float4*)(srcK + (size_t)row * H_ + col);
      *(float4*)&Vs[row * 196 + col] = *(const float4*)(srcV + (size_t)row * H_ + col);
    }
    if (kt + 1 < L2_ / 16) {
      __builtin_prefetch(srcK + (size_t)16 * H_ + (lane >> 1) * H_ + (lane & 1) * 96, 0, 0);
      __builtin_prefetch(srcV + (size_t)16 * H_ + (lane >> 1) * H_ + (lane & 1) * 96, 0, 0);
    }
    __syncthreads();

    v8f s = {};
#pragma unroll 4
    for (int dk = 0; dk < D_ / 4; ++dk) {
      v2f a  = lds_frag(Qs, 0, dk * 4, 196);
      v2f bf = lds_frag(Ks, 0, dk * 4, 196);
      s = wmma_f32(a, bf, s);
    }
    const bool msk = mb[kt * 16 + coln] != 0;
#pragma unroll
    for (int v = 0; v < 8; ++v) {
      const float sv = msk ? -__builtin_inff() : s[v];
      Ps[(v + 8 * half) * 20 + coln] = __expf(sv - m[v]);  // D-layout -> row-major
    }
    __syncthreads();
#pragma unroll
    for (int ks = 0; ks < 4; ++ks) {
      v2f a = lds_frag(Ps, 0, ks * 4, 20);                 // P as A-fragment
#pragma unroll
      for (int nt = 0; nt < 12; ++nt) {
        const int cv = nt * 16 + coln;
        v2f bf;
        bf.x = Vs[(ks * 4 + 2 * half) * 196 + cv];
        bf.y = Vs[(ks * 4 + 2 * half + 1) * 196 + cv];
        acc[nt] = wmma_f32(a, bf, acc[nt]);
      }
    }
  }

  // Epilogue: out[b, q, n*192 + d] = acc / l
  float* ob = out + ((size_t)b * L1_ + q0) * H_ + n * D_;
#pragma unroll
  for (int nt = 0; nt < 12; ++nt) {
#pragma unroll
    for (int v = 0; v < 8; ++v) {
      ob[(size_t)(v + 8 * half) * H_ + nt * 16 + coln] = acc[nt][v] / l[v];
    }
  }
}

// ---------------------------------------------------------------------------
extern "C" void kernel_launch(void* const* d_in, const int* in_sizes, int n_in,
                              void* d_out, int out_size, void* d_ws, size_t ws_size,
                              hipStream_t stream) {
  const float* x1_att = (const float*)d_in[0];
  const float* x2_att = (const float*)d_in[1];
  // d_in[2] (x1) is unused by the reference.
  const float* x2      = (const float*)d_in[3];
  const unsigned char* x2_mask = (const unsigned char*)d_in[4];  // bool array
  const float* W       = (const float*)d_in[5];
  const float* final_v = (const float*)d_in[6];
  float* out = (float*)d_out;

  // Workspace: projected keys (2 x 16384 x 768 fp32 = 100.7 MB)
  float* qkey = (float*)d_ws;
  float* kkey = qkey + (size_t)(B_ * L1_) * H_;

  const int gemm_blocks = (B_ * L1_ / 128) * (H_ / 64);  // 128 * 12
  key_gemm_relu<<<gemm_blocks, 128, 0, stream>>>(x1_att, W, final_v, 0, qkey);
  key_gemm_relu<<<gemm_blocks, 128, 0, stream>>>(x2_att, W, final_v, 1, kkey);

  const int attn_blocks = B_ * NL_ * (L1_ / 16);         // 4096 single-wave blocks
  attn_kernel<<<attn_blocks, 32, 0, stream>>>(qkey, kkey, x2, x2_mask, out);
}